// StickyHDPHMMVI_78683800863300
// MI455X (gfx1250) — compile-verified
//
#include <hip/hip_runtime.h>
#include <hip/hip_bf16.h>
#include <math.h>

#define Tn 8192
#define Kn 64
#define Dn 128

typedef float v2f __attribute__((ext_vector_type(2)));
typedef float v8f __attribute__((ext_vector_type(8)));

// ---------------- device helpers ----------------

__device__ __forceinline__ float digammaf_dev(float x){
  // recurrence lift + asymptotic series; inputs here are >= 0.1
  float r = 0.0f;
  while (x < 6.0f){ r -= 1.0f/x; x += 1.0f; }
  float xi = 1.0f/x;
  float x2 = xi*xi;
  return r + logf(x) - 0.5f*xi - x2*(0.0833333333f - x2*(0.00833333333f - x2*0.00396825397f));
}

__device__ __forceinline__ float wave_max32(float v){
  #pragma unroll
  for (int m=16;m>=1;m>>=1) v = fmaxf(v, __shfl_xor(v, m, 32));
  return v;
}
__device__ __forceinline__ float wave_sum32(float v){
  #pragma unroll
  for (int m=16;m>=1;m>>=1) v += __shfl_xor(v, m, 32);
  return v;
}

// ---------------- ElogA = digamma(phi) - digamma(rowsum) ----------------

__global__ void elogA_kernel(const float* __restrict__ phi, float* __restrict__ ElogA){
  int i = threadIdx.x;                 // 64 threads, one row each
  float s = 0.f;
  for (int j=0;j<Kn;++j) s += phi[i*Kn+j];
  float dr = digammaf_dev(s);
  for (int j=0;j<Kn;++j) ElogA[i*Kn+j] = digammaf_dev(phi[i*Kn+j]) - dr;
}

// ---------------- per-k: Cholesky, L^-1, U=sqrt(nu)L^-1, consts ----------------

__global__ void prep_kernel(const float* __restrict__ Psi, const float* __restrict__ Mu,
                            const float* __restrict__ Nu,
                            float* __restrict__ W, float* __restrict__ cvec,
                            float* __restrict__ ELamDiag, float* __restrict__ Ck){
  __shared__ float Lm[Dn*Dn];          // 64KB
  __shared__ float red[Dn];
  __shared__ float sm[Dn];
  __shared__ float s_invd, s_logdet, s_dsum, s_cn2;
  const int k = blockIdx.x;
  const int tid = threadIdx.x;         // 128 threads
  const float nu  = Nu[k];
  const float snu = sqrtf(nu);

  for (int idx=tid; idx<Dn*Dn; idx+=Dn) Lm[idx] = Psi[k*Dn*Dn + idx];
  sm[tid] = Mu[k*Dn + tid];
  __syncthreads();

  // in-place Cholesky (lower); thread = row
  for (int j=0;j<Dn;++j){
    float s = 0.f;
    if (tid >= j){
      s = Lm[tid*Dn+j];
      for (int p=0;p<j;++p) s -= Lm[tid*Dn+p]*Lm[j*Dn+p];
    }
    if (tid == j){
      float dg = sqrtf(s);
      Lm[j*Dn+j] = dg;
      s_invd = 1.0f/dg;
    }
    __syncthreads();
    if (tid > j) Lm[tid*Dn+j] = s * s_invd;
    __syncthreads();
  }

  // logdet(Psi) = 2*sum(log diag L)
  red[tid] = logf(Lm[tid*Dn+tid]);
  __syncthreads();
  if (tid==0){ float s=0.f; for (int i=0;i<Dn;++i) s+=red[i]; s_logdet = 2.0f*s; }
  __syncthreads();

  // sum_i digamma((nu+1-i)/2), i=1..D
  red[tid] = digammaf_dev((nu + 1.0f - (float)(tid+1))*0.5f);
  __syncthreads();
  if (tid==0){ float s=0.f; for (int i=0;i<Dn;++i) s+=red[i]; s_dsum = s; }
  __syncthreads();

  // in-place lower-triangular inversion: Lm <- L^-1 (row i sequential, cols parallel)
  for (int i=0;i<Dn;++i){
    float invd = 1.0f/Lm[i*Dn+i];
    float xv = 0.f;
    if (tid < i){
      float s = 0.f;
      for (int p=tid;p<i;++p) s += Lm[i*Dn+p]*Lm[p*Dn+tid];
      xv = -invd*s;
    } else if (tid == i){
      xv = invd;
    }
    __syncthreads();
    if (tid <= i) Lm[i*Dn+tid] = xv;
    __syncthreads();
  }

  // W[k][d][e] = snu * Linv[d][e] (explicit zeros above diagonal)
  for (int idx=tid; idx<Dn*Dn; idx+=Dn){
    int d = idx >> 7, e = idx & 127;
    W[k*Dn*Dn + idx] = (e <= d) ? snu*Lm[idx] : 0.f;
  }

  // c = U m ; ||c||^2
  float c = 0.f;
  for (int e=0;e<=tid;++e) c += Lm[tid*Dn+e]*sm[e];
  c *= snu;
  cvec[k*Dn+tid] = c;
  red[tid] = c*c;
  __syncthreads();
  if (tid==0){ float s=0.f; for (int i=0;i<Dn;++i) s+=red[i]; s_cn2 = s; }
  __syncthreads();

  // diag(E[Lambda])[d] = nu * sum_e Linv[e][d]^2
  float s2 = 0.f;
  for (int e=tid;e<Dn;++e){ float v = Lm[e*Dn+tid]; s2 += v*v; }
  ELamDiag[k*Dn+tid] = nu*s2;

  if (tid==0){
    float Elogdet = s_dsum + 128.0f*0.6931471805599453f - s_logdet;
    Ck[k] = 0.5f*Elogdet - 64.0f*1.8378770664093453f - 0.5f*s_cn2;
  }
}

// ---------------- fp32 WMMA GEMM: Q[t,k] = c_k.V - 0.5||V||^2, V = U_k x_t ----------------
// one wave per (16-row t-tile, k); triangular U -> skip e-tiles above diagonal

__global__ void __launch_bounds__(128) gemm_kernel(const float* __restrict__ X,
    const float* __restrict__ W, const float* __restrict__ cvec, float* __restrict__ Q){
  const int wv   = threadIdx.x >> 5;
  const int lane = threadIdx.x & 31;
  const int wid  = blockIdx.x*4 + wv;       // 32768 wave tasks
  const int k    = wid & 63;
  const int t0   = (wid >> 6) * 16;
  const int hi   = lane >> 4;
  const int l15  = lane & 15;
  const float* Xp = X + (t0 + l15)*Dn + 2*hi;            // A: rows m=l15, K pair by half
  const float* Wp = W + k*Dn*Dn + l15*Dn + 2*hi;         // B: cols n=l15 (row d0+n of U)
  float p1[8], p2[8];
  #pragma unroll
  for (int j=0;j<8;++j){ p1[j]=0.f; p2[j]=0.f; }
  #pragma unroll
  for (int dt=0;dt<8;++dt){
    v8f c = {0.f,0.f,0.f,0.f,0.f,0.f,0.f,0.f};
    const float* wb = Wp + dt*16*Dn;
    const int ebound = 16*(dt+1);           // U lower-triangular: e <= d only
    for (int e0=0;e0<ebound;e0+=4){
      v2f a = *(const v2f*)(Xp + e0);
      v2f b = *(const v2f*)(wb + e0);
      c = __builtin_amdgcn_wmma_f32_16x16x4_f32(false, a, false, b, (short)0, c, false, false);
    }
    float cc = cvec[k*Dn + dt*16 + l15];
    #pragma unroll
    for (int j=0;j<8;++j){
      float vv = c[j];                      // row m=j+8*hi, col d=dt*16+l15
      p1[j] += vv*cc;
      p2[j] += vv*vv;
    }
  }
  #pragma unroll
  for (int m=1;m<16;m<<=1){
    #pragma unroll
    for (int j=0;j<8;++j){
      p1[j] += __shfl_xor(p1[j], m, 32);
      p2[j] += __shfl_xor(p2[j], m, 32);
    }
  }
  #pragma unroll
  for (int j=0;j<8;++j){
    if (l15 == j) Q[(t0 + j + 8*hi)*Kn + k] = p1[j] - 0.5f*p2[j];
  }
}

// ---------------- finish logB: += Ck - 0.5 * diagv . ELamDiag ----------------

__global__ void finish_kernel(float* __restrict__ Q, const float* __restrict__ diagv,
                              const float* __restrict__ ELamDiag, const float* __restrict__ Ck){
  const int t = blockIdx.x, k = threadIdx.x;   // 64 threads
  __shared__ float dv[Dn];
  for (int d=k; d<Dn; d+=Kn) dv[d] = fmaxf(diagv[t*Dn+d], 0.f);
  __syncthreads();
  const float* ed = ELamDiag + k*Dn;
  float tr = 0.f;
  for (int d=0;d<Dn;++d) tr += dv[d]*ed[d];
  Q[t*Kn+k] += Ck[k] - 0.5f*tr;
}

// ---------------- sequential forward-backward: single wave, 2 states/lane ----------------

__global__ void fb_kernel(const float* __restrict__ logB, const float* __restrict__ ElogA,
                          const float* __restrict__ log_pi,
                          float* __restrict__ alpha, float* __restrict__ beta,
                          float* __restrict__ logZ, float* __restrict__ llout){
  __shared__ float E[4096];    // [i][j], forward reads column j = lane (conflict-free)
  __shared__ float ETs[4096];  // [j][i], backward reads i = lane (conflict-free)
  __shared__ float ap[64];
  __shared__ float bb[64];
  const int lane = threadIdx.x;  // 32 threads
  for (int idx=lane; idx<4096; idx+=32){
    float x = ElogA[idx];
    E[idx] = x;
    ETs[(idx & 63)*64 + (idx >> 6)] = x;
  }
  __syncthreads();
  const int j0 = lane, j1 = lane + 32;

  // t = 0
  float a0 = log_pi[j0] + logB[j0];
  float a1 = log_pi[j1] + logB[j1];
  float zm = wave_max32(fmaxf(a0,a1));
  float zs = wave_sum32(__expf(a0-zm)+__expf(a1-zm));
  float z  = zm + __logf(zs);
  a0 -= z; a1 -= z;
  alpha[j0]=a0; alpha[j1]=a1;
  ap[j0]=a0; ap[j1]=a1;
  if (lane==0) logZ[0]=z;
  float llsum = z;
  __syncthreads();

  // forward: ap normalized (<=0) and E in [-16,0) -> no per-j max pass needed
  for (int t=1;t<Tn;++t){
    float s0=0.f, s1=0.f;
    for (int i=0;i<64;++i){
      float api = ap[i];
      s0 += __expf(api + E[i*64+j0]);
      s1 += __expf(api + E[i*64+j1]);
    }
    float na0 = __logf(s0) + logB[t*Kn+j0];
    float na1 = __logf(s1) + logB[t*Kn+j1];
    zm = wave_max32(fmaxf(na0,na1));
    zs = wave_sum32(__expf(na0-zm)+__expf(na1-zm));
    z  = zm + __logf(zs);
    na0 -= z; na1 -= z;
    __syncthreads();
    ap[j0]=na0; ap[j1]=na1;
    alpha[t*Kn+j0]=na0; alpha[t*Kn+j1]=na1;
    if (lane==0) logZ[t]=z;
    llsum += z;
    __syncthreads();
  }
  if (lane==0) *llout = llsum;

  // backward
  beta[(Tn-1)*Kn+j0]=0.f; beta[(Tn-1)*Kn+j1]=0.f;
  __syncthreads();
  ap[j0]=0.f; ap[j1]=0.f;     // reuse ap as beta_next
  __syncthreads();
  for (int t=Tn-2;t>=0;--t){
    float v0 = logB[(t+1)*Kn+j0] + ap[j0];
    float v1 = logB[(t+1)*Kn+j1] + ap[j1];
    float mb = wave_max32(fmaxf(v0,v1));   // shift so exp args <= 0
    __syncthreads();
    bb[j0]=v0-mb; bb[j1]=v1-mb;
    __syncthreads();
    float zn = logZ[t+1];
    float s0=0.f, s1=0.f;
    for (int j=0;j<64;++j){
      float c = bb[j];
      s0 += __expf(ETs[j*64+j0] + c);
      s1 += __expf(ETs[j*64+j1] + c);
    }
    float nb0 = __logf(s0) + mb - zn;
    float nb1 = __logf(s1) + mb - zn;
    __syncthreads();
    ap[j0]=nb0; ap[j1]=nb1;
    beta[t*Kn+j0]=nb0; beta[t*Kn+j1]=nb1;
    __syncthreads();
  }
}

// ---------------- rhat = softmax(alpha+beta, axis=1) ----------------

__global__ void rhat_kernel(const float* __restrict__ alpha, const float* __restrict__ beta,
                            float* __restrict__ rhat){
  int t = blockIdx.x, k = threadIdx.x;   // 64 threads
  __shared__ float vsh[64];
  float x = alpha[t*Kn+k] + beta[t*Kn+k];
  vsh[k] = x;
  __syncthreads();
  float m = -3.0e38f;
  for (int j=0;j<64;++j) m = fmaxf(m, vsh[j]);
  float s = 0.f;
  for (int j=0;j<64;++j) s += __expf(vsh[j]-m);
  rhat[t*Kn+k] = __expf(x - m - __logf(s));
}

// ---------------- xihat: per-t self-normalized softmax over K*K ----------------

__global__ void xihat_kernel(const float* __restrict__ alpha, const float* __restrict__ beta,
                             const float* __restrict__ logB, const float* __restrict__ ElogA,
                             float* __restrict__ out){
  const int t = blockIdx.x;              // 0..T-2
  __shared__ float Es[4096];
  __shared__ float ar[64];
  __shared__ float br[64];
  __shared__ float redm[8];
  __shared__ float reds[8];
  const int tid  = threadIdx.x;          // 256 threads
  const int lane = tid & 31;
  const int wv   = tid >> 5;
  for (int idx=tid; idx<4096; idx+=256) Es[idx] = ElogA[idx];
  if (tid < 64) ar[tid] = alpha[t*Kn+tid];
  else if (tid < 128){ int j = tid-64; br[j] = logB[(t+1)*Kn+j] + beta[(t+1)*Kn+j]; }
  __syncthreads();
  float v[16];
  float m = -3.0e38f;
  #pragma unroll
  for (int p=0;p<16;++p){
    int idx = tid + 256*p;               // i = idx>>6, j = idx&63
    float x = ar[idx>>6] + Es[idx] + br[idx&63];
    v[p] = x; m = fmaxf(m,x);
  }
  m = wave_max32(m);
  if (lane==0) redm[wv]=m;
  __syncthreads();
  float M = redm[0];
  #pragma unroll
  for (int w=1;w<8;++w) M = fmaxf(M, redm[w]);
  float s = 0.f;
  #pragma unroll
  for (int p=0;p<16;++p) s += __expf(v[p]-M);
  s = wave_sum32(s);
  if (lane==0) reds[wv]=s;
  __syncthreads();
  float S = 0.f;
  #pragma unroll
  for (int w=0;w<8;++w) S += reds[w];
  float norm = M + __logf(S);
  float* op = out + (size_t)t*4096;
  #pragma unroll
  for (int p=0;p<16;++p) op[tid+256*p] = __expf(v[p]-norm);
}

// ---------------- launch ----------------

extern "C" void kernel_launch(void* const* d_in, const int* in_sizes, int n_in,
                              void* d_out, int out_size, void* d_ws, size_t ws_size,
                              hipStream_t stream){
  const float* mu_t     = (const float*)d_in[0];
  const float* diag_var = (const float*)d_in[1];
  const float* niw_mu   = (const float*)d_in[2];
  const float* niw_Psi  = (const float*)d_in[3];
  const float* niw_nu   = (const float*)d_in[4];
  const float* phi      = (const float*)d_in[5];
  const float* log_pi   = (const float*)d_in[6];

  char* ws = (char*)d_ws;
  float* W        = (float*)(ws);                         // 4,194,304 B
  float* cvec     = (float*)(ws + 4194304);               //    32,768 B
  float* ELamDiag = (float*)(ws + 4227072);               //    32,768 B
  float* Ck       = (float*)(ws + 4259840);               //       256 B
  float* ElogA    = (float*)(ws + 4260096);               //    16,384 B
  float* logB     = (float*)(ws + 4276480);               // 2,097,152 B
  float* alpha    = (float*)(ws + 6373632);               // 2,097,152 B
  float* beta     = (float*)(ws + 8470784);               // 2,097,152 B
  float* logZ     = (float*)(ws + 10567936);              //    32,768 B

  float* rhat  = (float*)d_out;
  float* xihat = rhat + (size_t)Tn*Kn;
  float* ll    = xihat + (size_t)(Tn-1)*Kn*Kn;

  elogA_kernel <<<1,    64,  0, stream>>>(phi, ElogA);
  prep_kernel  <<<64,   128, 0, stream>>>(niw_Psi, niw_mu, niw_nu, W, cvec, ELamDiag, Ck);
  gemm_kernel  <<<8192, 128, 0, stream>>>(mu_t, W, cvec, logB);
  finish_kernel<<<8192, 64,  0, stream>>>(logB, diag_var, ELamDiag, Ck);
  fb_kernel    <<<1,    32,  0, stream>>>(logB, ElogA, log_pi, alpha, beta, logZ, ll);
  rhat_kernel  <<<8192, 64,  0, stream>>>(alpha, beta, rhat);
  xihat_kernel <<<8191, 256, 0, stream>>>(alpha, beta, logB, ElogA, xihat);
}